// PointerNetwork_17841294147655
// MI455X (gfx1250) — compile-verified
//
#include <hip/hip_runtime.h>
#include <hip/hip_bf16.h>

typedef __attribute__((ext_vector_type(2))) float v2f;
typedef __attribute__((ext_vector_type(8))) float v8f;

static constexpr int B  = 32;
static constexpr int N  = 8192;
static constexpr int D  = 512;
static constexpr int K  = 256;

// ---------------------------------------------------------------------------
// Kernel A: scores[b,n] = dot(x[b,n,:], W) + bias   via V_WMMA_F32_16X16X4_F32
// One wave computes 16 rows. A = 16x4 f32 tile of x, B = W[k..k+3] broadcast
// into all 16 columns -> every column of the 16x16 accumulator holds the same
// 16 dot products. Memory-bound: streams x (512 MiB) exactly once.
// ---------------------------------------------------------------------------
__global__ __launch_bounds__(256) void scores_wmma_kernel(
    const float* __restrict__ x, const float* __restrict__ W,
    const float* __restrict__ bias, float* __restrict__ scores)
{
    __shared__ float sW[D];
    for (int i = threadIdx.x; i < D; i += blockDim.x) sW[i] = W[i];
    __syncthreads();

    const int lane = threadIdx.x & 31;
    const int wave = threadIdx.x >> 5;
    // 8 waves/block * 16 rows/wave = 128 rows/block; N/128 = 64 blocks per batch
    const int b       = blockIdx.x >> 6;
    const int rowBase = ((blockIdx.x & 63) << 7) + (wave << 4);

    // A layout: lanes 0-15 -> row = rowBase+lane, K = k..k+1 (VGPR0..1)
    //           lanes16-31 -> row = rowBase+lane-16, K = k+2..k+3
    const int koff = (lane >> 4) << 1;                 // 0 or 2
    const float* xrow = x + ((size_t)b * N + rowBase + (lane & 15)) * D + koff;

    v8f c0 = {}; v8f c1 = {};
    #pragma unroll 4
    for (int k = 0; k < D; k += 8) {
        v2f a0 = *(const v2f*)(xrow + k);
        v2f b0 = *(const v2f*)(&sW[k + koff]);         // W broadcast to all 16 cols
        c0 = __builtin_amdgcn_wmma_f32_16x16x4_f32(false, a0, false, b0,
                                                   (short)0, c0, false, false);
        v2f a1 = *(const v2f*)(xrow + k + 4);
        v2f b1 = *(const v2f*)(&sW[k + 4 + koff]);
        c1 = __builtin_amdgcn_wmma_f32_16x16x4_f32(false, a1, false, b1,
                                                   (short)0, c1, false, false);
    }
    v8f c = c0 + c1;

    // D layout: lanes 0-15 hold M=0..7 (VGPR0..7) at N=lane; lanes 16-31 hold M=8..15.
    // All columns identical -> lane 0 has rows 0..7, lane 16 has rows 8..15.
    const float bv = bias[0];
    if (lane == 0) {
        #pragma unroll
        for (int j = 0; j < 8; ++j)
            scores[(size_t)b * N + rowBase + j] = c[j] + bv;
    } else if (lane == 16) {
        #pragma unroll
        for (int j = 0; j < 8; ++j)
            scores[(size_t)b * N + rowBase + 8 + j] = c[j] + bv;
    }
}

// ---------------------------------------------------------------------------
// Kernel B: per-batch softmax + exact top-K radix select + bitonic sort.
// One 1024-thread block per batch; each thread owns 8 elements in registers.
// Keys = float bits of exp(s-max) (positive -> order-preserving as uint).
// ---------------------------------------------------------------------------
__global__ __launch_bounds__(1024) void softmax_topk_kernel(
    const float* __restrict__ scores, float* __restrict__ probs,
    int* __restrict__ topk_idx)
{
    __shared__ float red[32];
    __shared__ unsigned hist[256];
    __shared__ unsigned sh_prefix, sh_mask, sh_remK, sh_cnt;
    __shared__ unsigned long long pairs[K];

    const int b    = blockIdx.x;
    const int tid  = threadIdx.x;
    const int lane = tid & 31;
    const int wave = tid >> 5;
    const float* sc = scores + (size_t)b * N;

    // ---- pass 1: max ----
    float v[8];
    float m = -INFINITY;
    #pragma unroll
    for (int r = 0; r < 8; ++r) { v[r] = sc[tid + r * 1024]; m = fmaxf(m, v[r]); }
    #pragma unroll
    for (int off = 16; off > 0; off >>= 1) m = fmaxf(m, __shfl_down(m, off, 32));
    if (lane == 0) red[wave] = m;
    __syncthreads();
    if (tid == 0) { float mm = red[0]; for (int i = 1; i < 32; ++i) mm = fmaxf(mm, red[i]); red[0] = mm; }
    __syncthreads();
    m = red[0];
    __syncthreads();

    // ---- pass 2: exp + sum ----
    float s = 0.f;
    #pragma unroll
    for (int r = 0; r < 8; ++r) { v[r] = __expf(v[r] - m); s += v[r]; }
    #pragma unroll
    for (int off = 16; off > 0; off >>= 1) s += __shfl_down(s, off, 32);
    if (lane == 0) red[wave] = s;
    __syncthreads();
    if (tid == 0) { float ss = 0.f; for (int i = 0; i < 32; ++i) ss += red[i]; red[0] = ss; }
    __syncthreads();
    const float inv = 1.0f / red[0];
    #pragma unroll
    for (int r = 0; r < 8; ++r) probs[(size_t)b * N + tid + r * 1024] = v[r] * inv;

    // ---- radix select: exact K-th largest key (4 passes of 8 bits, MSB first) ----
    if (tid == 0) { sh_prefix = 0u; sh_mask = 0u; sh_remK = (unsigned)K; sh_cnt = 0u; }
    __syncthreads();
    for (int shift = 24; shift >= 0; shift -= 8) {
        if (tid < 256) hist[tid] = 0u;
        __syncthreads();
        const unsigned pfx = sh_prefix, msk = sh_mask;
        #pragma unroll
        for (int r = 0; r < 8; ++r) {
            unsigned key = __float_as_uint(v[r]);
            if ((key & msk) == pfx) atomicAdd(&hist[(key >> shift) & 255u], 1u);
        }
        __syncthreads();
        if (tid == 0) {
            unsigned cum = 0u, remK = sh_remK, dg = 0u;
            for (int bin = 255; bin >= 0; --bin) {
                if (cum + hist[bin] >= remK) { dg = (unsigned)bin; break; }
                cum += hist[bin];
            }
            sh_remK   = remK - cum;                 // still needed among == digit
            sh_prefix = pfx | (dg << shift);
            sh_mask   = msk | (255u << shift);
        }
        __syncthreads();
    }
    const unsigned T = sh_prefix;                   // exact threshold key

    // ---- compaction: all keys > T, then enough keys == T ----
    #pragma unroll
    for (int r = 0; r < 8; ++r) {
        const int i = tid + r * 1024;
        const unsigned key = __float_as_uint(v[r]);
        if (key > T) {
            unsigned pos = atomicAdd(&sh_cnt, 1u);
            pairs[pos] = ((unsigned long long)key << 32) | (unsigned)(N - 1 - i);
        }
    }
    __syncthreads();
    #pragma unroll
    for (int r = 0; r < 8; ++r) {
        const int i = tid + r * 1024;
        const unsigned key = __float_as_uint(v[r]);
        if (key == T) {
            unsigned pos = atomicAdd(&sh_cnt, 1u);
            if (pos < (unsigned)K)
                pairs[pos] = ((unsigned long long)key << 32) | (unsigned)(N - 1 - i);
        }
    }
    __syncthreads();

    // ---- bitonic sort, descending by (key, then ascending index) ----
    // packed low word is (N-1-i): descending packed => ascending index on ties.
    for (int ksz = 2; ksz <= K; ksz <<= 1) {
        for (int j = ksz >> 1; j > 0; j >>= 1) {
            const int i = tid;
            if (i < K) {
                const int ixj = i ^ j;
                if (ixj > i) {
                    unsigned long long a = pairs[i], bb = pairs[ixj];
                    const bool desc = ((i & ksz) == 0);
                    if (desc ? (a < bb) : (a > bb)) { pairs[i] = bb; pairs[ixj] = a; }
                }
            }
            __syncthreads();
        }
    }

    if (tid < K) {
        const unsigned long long p = pairs[tid];
        topk_idx[b * K + tid] = (int)(N - 1 - (unsigned)(p & 0xFFFFFFFFu));
    }
}

// ---------------------------------------------------------------------------
// Kernel C: gather selected rows. One wave per (b,k) row, float4 copies.
// ---------------------------------------------------------------------------
__global__ __launch_bounds__(256) void gather_kernel(
    const float* __restrict__ x, const int* __restrict__ topk_idx,
    float* __restrict__ selected)
{
    const int gwave = (int)((blockIdx.x * blockDim.x + threadIdx.x) >> 5);
    const int lane  = threadIdx.x & 31;
    if (gwave >= B * K) return;
    const int b   = gwave / K;
    const int row = topk_idx[gwave];
    const float4* src = (const float4*)(x + ((size_t)b * N + row) * D);
    float4*       dst = (float4*)(selected + (size_t)gwave * D);
    #pragma unroll
    for (int r = 0; r < 4; ++r) dst[lane + 32 * r] = src[lane + 32 * r];
}

extern "C" void kernel_launch(void* const* d_in, const int* in_sizes, int n_in,
                              void* d_out, int out_size, void* d_ws, size_t ws_size,
                              hipStream_t stream) {
    (void)in_sizes; (void)n_in; (void)out_size; (void)ws_size;
    const float* x    = (const float*)d_in[0];   // [B,N,D] f32
    const float* W    = (const float*)d_in[1];   // [1,D]   f32
    const float* bias = (const float*)d_in[2];   // [1]     f32

    float* out      = (float*)d_out;
    float* selected = out;                                        // B*K*D
    float* probs    = out + (size_t)B * K * D;                    // B*N
    int*   idx      = (int*)(out + (size_t)B * K * D + (size_t)B * N); // B*K int32

    float* scores = (float*)d_ws;                                 // B*N f32 scratch

    // A: WMMA GEMV — 8 waves/block * 16 rows = 128 rows/block; 64 blocks/batch
    scores_wmma_kernel<<<dim3(B * (N / 128)), dim3(256), 0, stream>>>(x, W, bias, scores);
    // B: per-batch softmax + exact top-K
    softmax_topk_kernel<<<dim3(B), dim3(1024), 0, stream>>>(scores, probs, idx);
    // C: gather B*K rows, one wave each
    gather_kernel<<<dim3((B * K * 32) / 256), dim3(256), 0, stream>>>(x, idx, selected);
}